// GCN_Net_52974126629470
// MI455X (gfx1250) — compile-verified
//
#include <hip/hip_runtime.h>
#include <hip/hip_bf16.h>

#define N_FEAT   256
#define HIDDEN   64
#define N_CLASS  2

typedef __attribute__((ext_vector_type(2))) float v2f;
typedef __attribute__((ext_vector_type(8))) float v8f;

// ---------------- degree / dinv ----------------
__global__ void k_init_deg(float* deg, int n) {
    int i = blockIdx.x * blockDim.x + threadIdx.x;
    if (i < n) deg[i] = 1.0f;                       // self-loop counts as 1
}

__global__ void k_count(const int* __restrict__ dst, float* deg, int e) {
    int i = blockIdx.x * blockDim.x + threadIdx.x;
    if (i < e) atomicAdd(&deg[dst[i]], 1.0f);
}

__global__ void k_dinv(float* deg, int n) {
    int i = blockIdx.x * blockDim.x + threadIdx.x;
    if (i < n) deg[i] = rsqrtf(deg[i]);             // in-place: deg -> dinv
}

// ---------------- GEMM1: h1 = x @ W1 (f32 WMMA), fused self-loop init ----
// One wave per 16-row strip computes the FULL 16x64 output (all 4 N-tiles):
// one A-fragment load feeds 4 WMMAs per K-step, so X streams from HBM once.
// M=100000=6250*16, N=64=4*16, K=256.
__global__ void k_gemm1_wmma(const float* __restrict__ X,
                             const float* __restrict__ W,
                             const float* __restrict__ dinv,
                             float* __restrict__ H,
                             float* __restrict__ AGG,
                             int mtiles) {
    const int mt   = blockIdx.x * (blockDim.x >> 5) + (threadIdx.x >> 5);
    if (mt >= mtiles) return;
    const int lane = threadIdx.x & 31;
    const int lr   = lane & 15;      // A: row within strip; B/C/D: col within tile
    const int half = lane >> 4;      // K sub-pair select (A/B), row-half (C/D)

    // A: lane holds X[mt*16+lr][k + 2*half + {0,1}]  (read once -> non-temporal)
    const float* xrow = X + (size_t)(mt * 16 + lr) * N_FEAT + 2 * half;
    // B: lane holds W[k + 2*half + {0,1}][nt*16+lr]
    const float* wb   = W + (size_t)(2 * half) * HIDDEN + lr;

    v8f c0 = {}, c1 = {}, c2 = {}, c3 = {};
#pragma unroll 4
    for (int k = 0; k < N_FEAT; k += 4) {
        v2f a;
        a.x = __builtin_nontemporal_load(xrow + k);
        a.y = __builtin_nontemporal_load(xrow + k + 1);
        const float* wk = wb + (size_t)k * HIDDEN;
        v2f b0, b1, b2, b3;
        b0.x = wk[0];       b0.y = wk[HIDDEN + 0];
        b1.x = wk[16];      b1.y = wk[HIDDEN + 16];
        b2.x = wk[32];      b2.y = wk[HIDDEN + 32];
        b3.x = wk[48];      b3.y = wk[HIDDEN + 48];
        c0 = __builtin_amdgcn_wmma_f32_16x16x4_f32(false, a, false, b0, (short)0, c0, false, false);
        c1 = __builtin_amdgcn_wmma_f32_16x16x4_f32(false, a, false, b1, (short)0, c1, false, false);
        c2 = __builtin_amdgcn_wmma_f32_16x16x4_f32(false, a, false, b2, (short)0, c2, false, false);
        c3 = __builtin_amdgcn_wmma_f32_16x16x4_f32(false, a, false, b3, (short)0, c3, false, false);
    }

    // D layout: VGPR v -> row (v + 8*half), col lr within each 16-wide tile
#pragma unroll
    for (int v = 0; v < 8; ++v) {
        const int row  = mt * 16 + v + 8 * half;
        const float di = dinv[row];
        const float dd = di * di;
        float* hr = H   + (size_t)row * HIDDEN + lr;
        float* ar = AGG + (size_t)row * HIDDEN + lr;
        float h0 = c0[v], h1 = c1[v], h2 = c2[v], h3 = c3[v];
        hr[0]  = h0;  ar[0]  = h0 * dd;   // pre-norm feats + self-loop init
        hr[16] = h1;  ar[16] = h1 * dd;
        hr[32] = h2;  ar[32] = h2 * dd;
        hr[48] = h3;  ar[48] = h3 * dd;
    }
}

// ---------------- edge aggregation layer 1 (thread per edge,feature) -----
__global__ void k_edge1(const int* __restrict__ src, const int* __restrict__ dst,
                        const float* __restrict__ dinv,
                        const float* __restrict__ H,
                        float* AGG, int e) {
    long long t = (long long)blockIdx.x * blockDim.x + threadIdx.x;
    int ei = (int)(t >> 6);          // / HIDDEN
    int f  = (int)(t & 63);
    if (ei >= e) return;
    int s = src[ei], d = dst[ei];
    float norm = dinv[s] * dinv[d];
    atomicAdd(&AGG[(size_t)d * HIDDEN + f], H[(size_t)s * HIDDEN + f] * norm);
}

// ---------------- layer1 bias+ReLU fused with GEMM2 + layer2 self-loop ---
__global__ void k_gemm2(const float* __restrict__ AGG1,
                        const float* __restrict__ b1,
                        const float* __restrict__ W2,
                        const float* __restrict__ dinv,
                        float* __restrict__ H2,
                        float* __restrict__ AGG2, int n) {
    int i = blockIdx.x * blockDim.x + threadIdx.x;
    if (i >= n) return;
    const float* r = AGG1 + (size_t)i * HIDDEN;
    float a0 = 0.f, a1 = 0.f;
#pragma unroll
    for (int f = 0; f < HIDDEN; ++f) {
        float v = r[f] + b1[f];
        v = v > 0.f ? v : 0.f;                      // ReLU of layer 1
        a0 += v * W2[2 * f];
        a1 += v * W2[2 * f + 1];
    }
    float di = dinv[i];
    float dd = di * di;
    H2[2 * i]     = a0;
    H2[2 * i + 1] = a1;
    AGG2[2 * i]     = a0 * dd;                      // self-loop init
    AGG2[2 * i + 1] = a1 * dd;
}

// ---------------- edge aggregation layer 2 (thread per edge) -------------
__global__ void k_edge2(const int* __restrict__ src, const int* __restrict__ dst,
                        const float* __restrict__ dinv,
                        const float* __restrict__ H2,
                        float* AGG2, int e) {
    int ei = blockIdx.x * blockDim.x + threadIdx.x;
    if (ei >= e) return;
    int s = src[ei], d = dst[ei];
    float norm = dinv[s] * dinv[d];
    atomicAdd(&AGG2[2 * d],     H2[2 * s]     * norm);
    atomicAdd(&AGG2[2 * d + 1], H2[2 * s + 1] * norm);
}

// ---------------- final bias + ReLU --------------------------------------
__global__ void k_final(const float* __restrict__ AGG2,
                        const float* __restrict__ b2,
                        float* __restrict__ out, int n) {
    int i = blockIdx.x * blockDim.x + threadIdx.x;
    if (i >= n) return;
    float v0 = AGG2[2 * i]     + b2[0];
    float v1 = AGG2[2 * i + 1] + b2[1];
    out[2 * i]     = v0 > 0.f ? v0 : 0.f;
    out[2 * i + 1] = v1 > 0.f ? v1 : 0.f;
}

extern "C" void kernel_launch(void* const* d_in, const int* in_sizes, int n_in,
                              void* d_out, int out_size, void* d_ws, size_t ws_size,
                              hipStream_t stream) {
    const float* x    = (const float*)d_in[0];
    const int*   esrc = (const int*)  d_in[1];
    const int*   edst = (const int*)  d_in[2];
    const float* W1   = (const float*)d_in[3];
    const float* b1   = (const float*)d_in[4];
    const float* W2   = (const float*)d_in[5];
    const float* b2   = (const float*)d_in[6];
    float* out = (float*)d_out;

    const int N = in_sizes[0] / N_FEAT;   // 100000
    const int E = in_sizes[1];            // 1200000

    // workspace layout (floats)
    float* ws   = (float*)d_ws;
    size_t off  = 0;
    float* dinv = ws + off; off += ((size_t)N + 63) & ~(size_t)63;
    float* H1   = ws + off; off += (size_t)N * HIDDEN;
    float* AGG1 = ws + off; off += (size_t)N * HIDDEN;
    float* H2   = ws + off; off += (size_t)N * N_CLASS;
    float* AGG2 = ws + off; off += (size_t)N * N_CLASS;

    const int TB = 256;

    k_init_deg<<<(N + TB - 1) / TB, TB, 0, stream>>>(dinv, N);
    k_count   <<<(E + TB - 1) / TB, TB, 0, stream>>>(edst, dinv, E);
    k_dinv    <<<(N + TB - 1) / TB, TB, 0, stream>>>(dinv, N);

    // GEMM1: one wave per 16-row strip (computes all of N=64); 8 waves/block
    const int mtiles = N / 16;                        // 6250
    k_gemm1_wmma<<<(mtiles + 7) / 8, TB, 0, stream>>>(x, W1, dinv, H1, AGG1, mtiles);

    const long long t1 = (long long)E * HIDDEN;       // 76.8M threads
    k_edge1<<<(unsigned)((t1 + TB - 1) / TB), TB, 0, stream>>>(esrc, edst, dinv, H1, AGG1, E);

    k_gemm2<<<(N + TB - 1) / TB, TB, 0, stream>>>(AGG1, b1, W2, dinv, H2, AGG2, N);
    k_edge2<<<(E + TB - 1) / TB, TB, 0, stream>>>(esrc, edst, dinv, H2, AGG2, E);
    k_final<<<(N + TB - 1) / TB, TB, 0, stream>>>(AGG2, b2, out, N);
}